// SSMLanguageModel_48438641164822
// MI455X (gfx1250) — compile-verified
//
#include <hip/hip_runtime.h>
#include <hip/hip_bf16.h>

#define D_MODEL 256
#define SEQ     8192
#define BATCH   8
#define TILE_T  256
#define NTILES  (SEQ / TILE_T)   // 32
#define CHUNK   16
#define ROWP    272              // padded LDS row stride in floats (t + (t>>4) swizzle, max 270)
#define KROW    32               // kappa row: [0..15]=0 padding, [16..31]=kappa[0..15]

typedef __attribute__((ext_vector_type(2))) float v2f;
typedef __attribute__((ext_vector_type(8))) float v8f;

struct Cx { float r, i; };

__device__ __forceinline__ Cx cmul(Cx a, Cx b) {
    Cx o; o.r = a.r * b.r - a.i * b.i; o.i = a.r * b.i + a.i * b.r; return o;
}
// a*h + add  (full complex fma)
__device__ __forceinline__ Cx cfma(Cx a, Cx h, Cx add) {
    Cx o; o.r = a.r * h.r - a.i * h.i + add.r; o.i = a.r * h.i + a.i * h.r + add.i; return o;
}
// base decay a = exp(lm) * (cos ph + i sin ph)  (single accurate sincos per (d,s))
__device__ __forceinline__ Cx cbase(float lm, float ph) {
    float s, c; sincosf(ph, &s, &c);
    float e = expf(lm);
    Cx o; o.r = e * c; o.i = e * s; return o;
}
// a^(2^k) by repeated squaring
__device__ __forceinline__ Cx csqr_n(Cx a, int k) {
    #pragma unroll
    for (int j = 0; j < 8; ++j) { if (j < k) a = cmul(a, a); }
    return a;
}
__device__ __forceinline__ Cx shup16(Cx v, unsigned d) {
    Cx o; o.r = __shfl_up(v.r, d, 16); o.i = __shfl_up(v.i, d, 16); return o;
}

// ---------------------------------------------------------------------------
// K1: per (b, tile, d) local end-state over 256 steps (zero-init recurrence)
// ws[(b*NTILES+tile)*D_MODEL + d] = {h0.r, h0.i, h1.r, h1.i}
// ---------------------------------------------------------------------------
__global__ __launch_bounds__(256) void ssm_k1_tile_end(
    const float* __restrict__ x, const float* __restrict__ lam, const float* __restrict__ aph,
    const float* __restrict__ Bre, const float* __restrict__ Bim, float4* __restrict__ ws)
{
    int tid  = blockIdx.x * 256 + threadIdx.x;     // 65536 = BATCH*NTILES*D_MODEL
    int d    = tid & (D_MODEL - 1);
    int bt   = tid >> 8;                           // b*NTILES + tile
    int tile = bt & (NTILES - 1);
    int b    = bt >> 5;

    Cx a0 = cbase(lam[d * 2 + 0], aph[d * 2 + 0]);
    Cx a1 = cbase(lam[d * 2 + 1], aph[d * 2 + 1]);
    Cx B0; B0.r = Bre[d * 2 + 0]; B0.i = Bim[d * 2 + 0];
    Cx B1; B1.r = Bre[d * 2 + 1]; B1.i = Bim[d * 2 + 1];

    const float* xp = x + ((size_t)b * SEQ + (size_t)tile * TILE_T) * D_MODEL + d;
    Cx h0; h0.r = 0.f; h0.i = 0.f;
    Cx h1; h1.r = 0.f; h1.i = 0.f;
    for (int t = 0; t < TILE_T; ++t) {
        float xv = xp[(size_t)t * D_MODEL];
        Cx bx0; bx0.r = B0.r * xv; bx0.i = B0.i * xv;
        Cx bx1; bx1.r = B1.r * xv; bx1.i = B1.i * xv;
        h0 = cfma(a0, h0, bx0);
        h1 = cfma(a1, h1, bx1);
    }
    ws[tid] = make_float4(h0.r, h0.i, h1.r, h1.i);
}

// ---------------------------------------------------------------------------
// K2: serial scan over the 32 tiles per (b,d); in-place: slot becomes the
// EXCLUSIVE carry (state entering that tile). a^256 via 8 squarings.
// ---------------------------------------------------------------------------
__global__ __launch_bounds__(256) void ssm_k2_scan(
    const float* __restrict__ lam, const float* __restrict__ aph, float4* __restrict__ ws)
{
    int tid = blockIdx.x * 256 + threadIdx.x;      // 2048 = BATCH*D_MODEL
    int d   = tid & (D_MODEL - 1);
    int b   = tid >> 8;

    Cx a256_0 = csqr_n(cbase(lam[d * 2 + 0], aph[d * 2 + 0]), 8);
    Cx a256_1 = csqr_n(cbase(lam[d * 2 + 1], aph[d * 2 + 1]), 8);

    Cx c0; c0.r = 0.f; c0.i = 0.f;
    Cx c1; c1.r = 0.f; c1.i = 0.f;
    for (int tile = 0; tile < NTILES; ++tile) {
        int idx = (b * NTILES + tile) * D_MODEL + d;
        float4 e = ws[idx];
        ws[idx] = make_float4(c0.r, c0.i, c1.r, c1.i);  // exclusive carry into this tile
        Cx e0; e0.r = e.x; e0.i = e.y;
        Cx e1; e1.r = e.z; e1.i = e.w;
        c0 = cfma(a256_0, c0, e0);
        c1 = cfma(a256_1, c1, e1);
    }
}

// ---------------------------------------------------------------------------
// K3: main kernel. Block = 512 threads = 16 waves; wave w handles channel
// dbase+w over one 256-step tile of one batch. Triangular-Toeplitz 16x16
// conv via 4x V_WMMA_F32_16X16X4_F32 (columns = 16 chunks), carry correction
// via wave-level scan, skip term, coalesced LDS-staged I/O.
// ---------------------------------------------------------------------------
__global__ __launch_bounds__(512) void ssm_k3_conv(
    const float* __restrict__ x, const float* __restrict__ lam, const float* __restrict__ aph,
    const float* __restrict__ Bre, const float* __restrict__ Bim,
    const float* __restrict__ Cre, const float* __restrict__ Cim,
    const float* __restrict__ Dd, const float4* __restrict__ carry, float* __restrict__ y)
{
    __shared__ float xt[16 * ROWP];     // [channel-in-group][swizzled t]
    __shared__ float kap[16 * KROW];    // [channel-in-group][zero-pad | kernel taps]

    int tid     = threadIdx.x;
    int blk     = blockIdx.x;           // 4096 = 16 dgrp * BATCH * NTILES
    int dgrp    = blk & 15;
    int lin     = blk >> 4;             // b*NTILES + tileIdx
    int tileIdx = lin & (NTILES - 1);
    int b       = lin >> 5;
    int dbase   = dgrp * 16;
    int t0      = tileIdx * TILE_T;

    // -- stage x tile (coalesced 64B runs) --
    const float* xb = x + ((size_t)b * SEQ + t0) * D_MODEL + dbase;
    for (int i = tid; i < TILE_T * 16; i += 512) {
        int dl = i & 15, t = i >> 4;
        xt[dl * ROWP + t + (t >> 4)] = xb[(size_t)t * D_MODEL + dl];
    }

    int lane = tid & 31;
    int w    = tid >> 5;        // 0..15 -> channel offset within group
    int d    = dbase + w;
    int half = lane >> 4;
    int n    = lane & 15;       // column / chunk index, also A-row index

    Cx a0 = cbase(lam[d * 2 + 0], aph[d * 2 + 0]);
    Cx a1 = cbase(lam[d * 2 + 1], aph[d * 2 + 1]);
    Cx B0; B0.r = Bre[d * 2 + 0]; B0.i = Bim[d * 2 + 0];
    Cx B1; B1.r = Bre[d * 2 + 1]; B1.i = Bim[d * 2 + 1];
    Cx C0; C0.r = Cre[d * 2 + 0]; C0.i = Cim[d * 2 + 0];
    Cx C1; C1.r = Cre[d * 2 + 1]; C1.i = Cim[d * 2 + 1];
    Cx cb0 = cmul(C0, B0);
    Cx cb1 = cmul(C1, B1);

    // power ladders: a^2, a^4, a^8, a^16 by squaring
    Cx a8_0 = csqr_n(a0, 3), a8_1 = csqr_n(a1, 3);
    Cx a16_0 = cmul(a8_0, a8_0), a16_1 = cmul(a8_1, a8_1);

    // -- kappa[m] = Re(CB0*a0^m) + Re(CB1*a1^m); lane n keeps kappa[n] --
    Cx p0; p0.r = 1.f; p0.i = 0.f;
    Cx p1; p1.r = 1.f; p1.i = 0.f;
    float kv = 0.f;
    #pragma unroll
    for (int m = 0; m < CHUNK; ++m) {
        float km = cb0.r * p0.r - cb0.i * p0.i + cb1.r * p1.r - cb1.i * p1.i;
        if (m == n) kv = km;
        p0 = cmul(p0, a0);
        p1 = cmul(p1, a1);
    }
    // zero-padded kappa row: indices 0..15 = 0, 16..31 = kappa[0..15]
    if (lane < 16) {
        kap[w * KROW + lane]      = 0.f;
        kap[w * KROW + 16 + lane] = kv;
    }
    __syncthreads();

    // -- 16x16 triangular-Toeplitz conv: 4 chained f32 WMMAs (K=4 each) --
    // A layout: lanes0-15 M=lane (K = 4p+{0,1}), lanes16-31 M=lane-16 (K = 4p+{2,3})
    // B layout: lanes0-15 N=lane row K=4p+{0,1}, lanes16-31 N=lane-16 row K=4p+{2,3}
    // A value = kappa[n-k], read unconditionally through the zero-padded row.
    v8f acc = {};
    const int kbase = w * KROW + 16 + n;   // points at kappa[n]
    #pragma unroll
    for (int p = 0; p < 4; ++p) {
        int k0 = p * 4 + half * 2;
        v2f A, Bv;
        A.x = kap[kbase - k0];
        A.y = kap[kbase - k0 - 1];
        int tb0 = n * 16 + k0;
        int tb1 = tb0 + 1;
        Bv.x = xt[w * ROWP + tb0 + (tb0 >> 4)];
        Bv.y = xt[w * ROWP + tb1 + (tb1 >> 4)];
        acc = __builtin_amdgcn_wmma_f32_16x16x4_f32(
            false, A, false, Bv, (short)0, acc, false, false);
    }

    // -- per-chunk local end state for column n: h_end = B * g, g = sum a^{15-k} x_k --
    Cx g0; g0.r = 0.f; g0.i = 0.f;
    Cx g1; g1.r = 0.f; g1.i = 0.f;
    #pragma unroll
    for (int k = 0; k < CHUNK; ++k) {
        int t = n * 16 + k;
        float xv = xt[w * ROWP + t + (t >> 4)];
        Cx xr; xr.r = xv; xr.i = 0.f;
        g0 = cfma(a0, g0, xr);
        g1 = cfma(a1, g1, xr);
    }
    Cx E0 = cmul(B0, g0), E1 = cmul(B1, g1);

    // -- inclusive Hillis-Steele scan across 16 chunks: segment = (P=a^16, V=E) --
    Cx P0 = a16_0, P1 = a16_1, V0 = E0, V1 = E1;
    #pragma unroll
    for (int j = 1; j < 16; j <<= 1) {
        Cx oP0 = shup16(P0, j), oV0 = shup16(V0, j);
        Cx oP1 = shup16(P1, j), oV1 = shup16(V1, j);
        if (n >= j) {
            V0 = cfma(P0, oV0, V0); P0 = cmul(P0, oP0);
            V1 = cfma(P1, oV1, V1); P1 = cmul(P1, oP1);
        }
    }
    // exclusive: carry entering chunk n
    Cx eP0 = shup16(P0, 1), eV0 = shup16(V0, 1);
    Cx eP1 = shup16(P1, 1), eV1 = shup16(V1, 1);
    if (n == 0) {
        eP0.r = 1.f; eP0.i = 0.f; eV0.r = 0.f; eV0.i = 0.f;
        eP1.r = 1.f; eP1.i = 0.f; eV1.r = 0.f; eV1.i = 0.f;
    }
    float4 c0v = carry[(b * NTILES + tileIdx) * D_MODEL + d];  // exclusive tile carry
    Cx t0c; t0c.r = c0v.x; t0c.i = c0v.y;
    Cx t1c; t1c.r = c0v.z; t1c.i = c0v.w;
    Cx cin0 = cfma(eP0, t0c, eV0);
    Cx cin1 = cfma(eP1, t1c, eV1);

    // -- outputs: D rows M = v + half*8, col N = n; y = conv + Re(C a^{M+1} cin) + D*x --
    // u = C * a^{M+1}, starting at M+1 = half*8 + 1 (a^9 = a^8 * a), stepped by a.
    float dD = Dd[d];
    Cx s0 = (half ? cmul(a8_0, a0) : a0);
    Cx s1 = (half ? cmul(a8_1, a1) : a1);
    Cx u0 = cmul(C0, s0);
    Cx u1 = cmul(C1, s1);
    float yv[8];
    #pragma unroll
    for (int v = 0; v < 8; ++v) {
        int M  = v + half * 8;
        int ts = n * 16 + M;
        float xs = xt[w * ROWP + ts + (ts >> 4)];
        float corr = u0.r * cin0.r - u0.i * cin0.i + u1.r * cin1.r - u1.i * cin1.i;
        yv[v] = acc[v] + corr + dD * xs;
        u0 = cmul(u0, a0);
        u1 = cmul(u1, a1);
    }

    // -- stage y through LDS and store coalesced --
    __syncthreads();
    #pragma unroll
    for (int v = 0; v < 8; ++v) {
        int M  = v + half * 8;
        int ts = n * 16 + M;
        xt[w * ROWP + ts + (ts >> 4)] = yv[v];
    }
    __syncthreads();
    float* yb = y + ((size_t)b * SEQ + t0) * D_MODEL + dbase;
    for (int i = tid; i < TILE_T * 16; i += 512) {
        int dl = i & 15, t = i >> 4;
        yb[(size_t)t * D_MODEL + dl] = xt[dl * ROWP + t + (t >> 4)];
    }
}

extern "C" void kernel_launch(void* const* d_in, const int* in_sizes, int n_in,
                              void* d_out, int out_size, void* d_ws, size_t ws_size,
                              hipStream_t stream)
{
    const float* x   = (const float*)d_in[0];
    const float* lam = (const float*)d_in[1];
    const float* aph = (const float*)d_in[2];
    const float* Bre = (const float*)d_in[3];
    const float* Bim = (const float*)d_in[4];
    const float* Cre = (const float*)d_in[5];
    const float* Cim = (const float*)d_in[6];
    const float* Dd  = (const float*)d_in[7];
    float*  y  = (float*)d_out;
    float4* ws = (float4*)d_ws;   // needs BATCH*NTILES*D_MODEL*16B = 1 MB

    ssm_k1_tile_end<<<(BATCH * NTILES * D_MODEL) / 256, 256, 0, stream>>>(x, lam, aph, Bre, Bim, ws);
    ssm_k2_scan<<<(BATCH * D_MODEL) / 256, 256, 0, stream>>>(lam, aph, ws);
    ssm_k3_conv<<<16 * BATCH * NTILES, 512, 0, stream>>>(x, lam, aph, Bre, Bim, Cre, Cim, Dd, ws, y);
}